// KnowledgeLayer_46059229282759
// MI455X (gfx1250) — compile-verified
//
#include <hip/hip_runtime.h>
#include <hip/hip_bf16.h>

#define BATCH 512
#define N_VARS 4096
#define LOG2F_C 0.6931471805599453f
#define EPSF 1e-15f

// ---------------------------------------------------------------------------
// CDNA5 async global->LDS staging helpers (gfx1250 ASYNCcnt path)
// Builtin signature (verified via hipcc diagnostic in a prior round):
//   (v4i addrspace(1)* global_src, v4i addrspace(3)* lds_dst, imm offset, imm cpol)
// ---------------------------------------------------------------------------
#if defined(__HIP_DEVICE_COMPILE__) && __has_builtin(__builtin_amdgcn_global_load_async_to_lds_b128)
#define USE_ASYNC_LDS 1
#else
#define USE_ASYNC_LDS 0
#endif

typedef int v4i __attribute__((vector_size(16)));
#if USE_ASYNC_LDS
typedef __attribute__((address_space(1))) v4i as1_v4i;
typedef __attribute__((address_space(3))) v4i as3_v4i;
#endif

__device__ __forceinline__ void stage16(const float* __restrict__ g, float* __restrict__ l) {
#if USE_ASYNC_LDS
    __builtin_amdgcn_global_load_async_to_lds_b128(
        (as1_v4i*)g, (as3_v4i*)l, /*offset=*/0, /*cpol=*/0);
#else
    float4 v = *(const float4*)g;
    *(float4*)l = v;
#endif
}

__device__ __forceinline__ void async_wait_all() {
#if USE_ASYNC_LDS
#if __has_builtin(__builtin_amdgcn_s_wait_asynccnt)
    __builtin_amdgcn_s_wait_asynccnt(0);
#else
    asm volatile("s_wait_asynccnt 0" ::: "memory");
#endif
#endif
}

// ---------------------------------------------------------------------------
// Product layer, fanin 4.  One block (512 thr = 16 wave32) per output row.
// 4 gathered rows x 2048B staged into LDS: each lane issues one async b128.
// Row pointer is wave-uniform (128-thread groups = 4 wave32s per row), so
// hoist it to an SGPR with readfirstlane.
// ---------------------------------------------------------------------------
__global__ void product4_kernel(const float* __restrict__ hin,
                                const int* __restrict__ ptrs,
                                float* __restrict__ hout) {
    __shared__ float smem[4 * BATCH];
    const int j = blockIdx.x;
    const int t = threadIdx.x;  // 0..511
    {
        const int r = t >> 7;            // which of the 4 gathered rows
        const int c = (t & 127) << 2;    // float offset within row (16B chunks)
        const int p = __builtin_amdgcn_readfirstlane(ptrs[(size_t)j * 4 + r]);
        stage16(hin + (size_t)p * BATCH + c, &smem[r * BATCH + c]);
    }
    async_wait_all();
    __syncthreads();
    float s = smem[t] + smem[BATCH + t] + smem[2 * BATCH + t] + smem[3 * BATCH + t];
    hout[(size_t)j * BATCH + t] = s;
}

// ---------------------------------------------------------------------------
// Sum layer, fanin 2 (logsumexp).  One block per output row.
// 2 rows x 2048B staged to LDS by the first 8 waves (256 lanes x b128).
// ---------------------------------------------------------------------------
__global__ void sum2_kernel(const float* __restrict__ hin,
                            const int* __restrict__ ptrs,
                            float* __restrict__ hout) {
    __shared__ float smem[2 * BATCH];
    const int j = blockIdx.x;
    const int t = threadIdx.x;  // 0..511
    if (t < 256) {
        const int r = t >> 7;
        const int c = (t & 127) << 2;
        const int p = __builtin_amdgcn_readfirstlane(ptrs[(size_t)j * 2 + r]);
        stage16(hin + (size_t)p * BATCH + c, &smem[r * BATCH + c]);
    }
    async_wait_all();
    __syncthreads();
    float g0 = smem[t];
    float g1 = smem[BATCH + t];
    float a  = fmaxf(g0, g1);
    float s  = expf(g0 - a) + expf(g1 - a);
    hout[(size_t)j * BATCH + t] = logf(s + EPSF) + a;
}

// ---------------------------------------------------------------------------
// encode_input.  h[0][:] = -inf, h[1][:] = 0,
// h[2+2i][:] = x[i], h[3+2i][:] = log1mexp(x[i]).
// ---------------------------------------------------------------------------
__global__ void encode_kernel(const float* __restrict__ x, float* __restrict__ h) {
    int tid = blockIdx.x * blockDim.x + threadIdx.x;   // (N_VARS+1)*BATCH threads
    if (tid < BATCH) {
        h[tid] = -INFINITY;
        h[BATCH + tid] = 0.0f;
        return;
    }
    int idx = tid - BATCH;
    if (idx >= N_VARS * BATCH) return;
    float xv = x[idx];
    int i = idx >> 9;           // var index (BATCH = 512)
    int b = idx & (BATCH - 1);  // batch column
    // accurate log(1 - exp(xv)), xv < 0
    float neg = (xv > -LOG2F_C) ? logf(-expm1f(xv)) : log1pf(-expf(xv));
    size_t base = (size_t)(2 + 2 * i) * BATCH + b;
    h[base]         = xv;
    h[base + BATCH] = neg;
}

// ---------------------------------------------------------------------------
// Launch: encode -> prod(16384,f4) -> sum(8192,f2) -> prod(4096,f4) -> sum(2048,f2)
// Workspace: [h0: 8194*512 f32 | h1: 16384*512 f32], then h2 reuses h0's slot,
// h3 reuses h1's slot. Peak usage ~48 MB (all layers are L2-resident on 192MB L2).
// ---------------------------------------------------------------------------
extern "C" void kernel_launch(void* const* d_in, const int* in_sizes, int n_in,
                              void* d_out, int out_size, void* d_ws, size_t ws_size,
                              hipStream_t stream) {
    (void)in_sizes; (void)n_in; (void)out_size; (void)ws_size;

    const float* x  = (const float*)d_in[0];
    const int*   p0 = (const int*)d_in[1];   // ptrs0 (seg arrays are the implicit repeat pattern)
    const int*   p1 = (const int*)d_in[3];   // ptrs1
    const int*   p2 = (const int*)d_in[5];   // ptrs2
    const int*   p3 = (const int*)d_in[7];   // ptrs3

    char*  ws = (char*)d_ws;
    float* h0 = (float*)ws;                              // (2*4096+2)*512 f32 = 16,781,312 B
    float* h1 = (float*)(ws + (size_t)8194 * BATCH * 4); // 16384*512 f32 = 33,554,432 B
    float* h2 = h0;                                      // 8192*512 f32 (fits in h0 slot)
    float* h3 = h1;                                      // 4096*512 f32 (fits in h1 slot)
    float* out = (float*)d_out;                          // 2048*512 f32

    encode_kernel  <<<(N_VARS + 1) * BATCH / 256, 256, 0, stream>>>(x, h0);
    product4_kernel<<<16384, BATCH, 0, stream>>>(h0, p0, h1);
    sum2_kernel    <<< 8192, BATCH, 0, stream>>>(h1, p1, h2);
    product4_kernel<<< 4096, BATCH, 0, stream>>>(h2, p2, h3);
    sum2_kernel    <<< 2048, BATCH, 0, stream>>>(h3, p3, out);
}